// KBAM3D_26061861552431
// MI455X (gfx1250) — compile-verified
//
#include <hip/hip_runtime.h>
#include <hip/hip_bf16.h>
#include <stdint.h>

typedef __attribute__((ext_vector_type(2))) float v2f;
typedef __attribute__((ext_vector_type(8))) float v8f;
typedef __attribute__((ext_vector_type(4))) unsigned int u32x4;
typedef __attribute__((ext_vector_type(8))) int i32x8;
typedef __attribute__((ext_vector_type(4))) int i32x4;

#define C_CH   128
#define R_CH   8
#define NB     8            // GRID_SIZE + SPLINE_ORDER
#define HWD    262144       // 64*64*64
#define TILE_V 128

__device__ __forceinline__ float sigmoidf_(float v) {
    return __builtin_amdgcn_rcpf(1.0f + __expf(-v));   // v_rcp_f32, no div chain
}
__device__ __forceinline__ float siluf_(float v) { return v * sigmoidf_(v); }

// Cox-de Boor degree-3 B-spline bases on uniform grid (GRID_SIZE=5 -> 8 bases).
// Same recursion as the reference; knot gaps are the constant j*h, so we
// multiply by compile-time reciprocals instead of emitting IEEE div sequences.
__device__ __forceinline__ void bspl8(float x, float* out) {
    const float h = 2.0f / 5.0f;
    float grid[12];
#pragma unroll
    for (int t = 0; t < 12; ++t) grid[t] = (float)(t - 3) * h - 1.0f;
    float bb[11];
#pragma unroll
    for (int i = 0; i < 11; ++i)
        bb[i] = (x >= grid[i] && x < grid[i + 1]) ? 1.0f : 0.0f;
#pragma unroll
    for (int j = 1; j <= 3; ++j) {
        const float rinv = 1.0f / ((float)j * h);      // constant-folded
#pragma unroll
        for (int i = 0; i < 11; ++i) {
            if (i + j < 11) {
                bb[i] = (x - grid[i]) * rinv * bb[i]
                      + (grid[i + j + 1] - x) * rinv * bb[i + 1];
            }
        }
    }
#pragma unroll
    for (int i = 0; i < NB; ++i) out[i] = bb[i];
}

// ---------------------------------------------------------------------------
// Kernel 1: per-(b,c) avg+max pooling over 64^3 spatial elements.
// ---------------------------------------------------------------------------
__global__ void kbam_pool_kernel(const float* __restrict__ x,
                                 float* __restrict__ yavg,
                                 float* __restrict__ ymax) {
    __shared__ float ssum[256];
    __shared__ float smax[256];
    const int bc  = blockIdx.x;                 // 0..255 = b*128+c
    const int tid = threadIdx.x;                // 256 threads
    const float4* p = (const float4*)(x + (size_t)bc * HWD);

    float s = 0.0f;
    float m = -3.402823466e+38f;
    for (int i = tid; i < HWD / 4; i += 256) {
        __builtin_prefetch((const void*)(p + i + 1024), 0, 0);  // global_prefetch
        float4 v = p[i];
        s += (v.x + v.y) + (v.z + v.w);
        m = fmaxf(m, fmaxf(fmaxf(v.x, v.y), fmaxf(v.z, v.w)));
    }
    ssum[tid] = s;
    smax[tid] = m;
    __syncthreads();
#pragma unroll
    for (int off = 128; off > 0; off >>= 1) {
        if (tid < off) {
            ssum[tid] += ssum[tid + off];
            smax[tid] = fmaxf(smax[tid], smax[tid + off]);
        }
        __syncthreads();
    }
    if (tid == 0) {
        yavg[bc] = ssum[0] * (1.0f / (float)HWD);
        ymax[bc] = smax[0];
    }
}

// ---------------------------------------------------------------------------
// Kernel 2: channel attention KAN MLP + precompute of the per-segment cubic
// coefficients of the spatial 1->1 spline:
//   sum_g B_g(s)*sks[g] on segment m (knots t_m=(m-3)*h-1, u=(s-t_m)/h) is
//   c0 + c1*u + c2*u^2 + c3*u^3 with the uniform B-spline matrix.
// ---------------------------------------------------------------------------
__global__ void kbam_chatt_kernel(const float* __restrict__ yavg,
                                  const float* __restrict__ ymax,
                                  const float* __restrict__ ck1_base,    // [8,128]
                                  const float* __restrict__ ck1_spline,  // [8,128,8]
                                  const float* __restrict__ ck2_base,    // [128,8]
                                  const float* __restrict__ ck2_spline,  // [128,8,8]
                                  const float* __restrict__ conv_w,      // [128]
                                  const float* __restrict__ sk_spline,   // [8]
                                  float* __restrict__ ca_out,
                                  float* __restrict__ caw_out,
                                  float4* __restrict__ coeff_out) {      // [11]
    __shared__ float red[2 * R_CH * C_CH];   // [(b*8+o)*128 + c]
    __shared__ float h1s[2 * R_CH];          // silu(relu(h1))
    __shared__ float h1b[2 * R_CH * NB];     // bspl(relu(h1))
    const int tid = threadIdx.x;             // 256
    const int b = tid >> 7;
    const int c = tid & 127;

    // ---- spatial-spline segment coefficients (11 segments over [-2.2, 2.2))
    if (tid < 11) {
        float S[4];
#pragma unroll
        for (int r = 0; r < 4; ++r) {
            const int g = tid - 3 + r;
            S[r] = (g >= 0 && g < NB) ? sk_spline[g] : 0.0f;
        }
        const float c0 = (S[0] + 4.0f * S[1] + S[2]) * (1.0f / 6.0f);
        const float c1 = (S[2] - S[0]) * 0.5f;
        const float c2 = (S[0] - 2.0f * S[1] + S[2]) * 0.5f;
        const float c3 = (S[3] - S[0] + 3.0f * (S[1] - S[2])) * (1.0f / 6.0f);
        coeff_out[tid] = make_float4(c0, c1, c2, c3);
    }

    float acc2 = 0.0f;
    for (int src = 0; src < 2; ++src) {
        const float y = (src == 0 ? yavg : ymax)[b * C_CH + c];
        const float sy = siluf_(y);
        float bs[NB];
        bspl8(y, bs);
#pragma unroll
        for (int o = 0; o < R_CH; ++o) {
            float v = sy * ck1_base[o * C_CH + c];
#pragma unroll
            for (int g = 0; g < NB; ++g)
                v += bs[g] * ck1_spline[(o * C_CH + c) * NB + g];
            red[(b * R_CH + o) * C_CH + c] = v;
        }
        __syncthreads();
        if (tid < 2 * R_CH) {
            float hsum = 0.0f;
            for (int i = 0; i < C_CH; ++i) hsum += red[tid * C_CH + i];
            hsum = fmaxf(hsum, 0.0f);        // relu
            h1s[tid] = siluf_(hsum);
            bspl8(hsum, &h1b[tid * NB]);
        }
        __syncthreads();
        float o2 = 0.0f;
#pragma unroll
        for (int r = 0; r < R_CH; ++r) {
            o2 += h1s[b * R_CH + r] * ck2_base[c * R_CH + r];
#pragma unroll
            for (int g = 0; g < NB; ++g)
                o2 += h1b[(b * R_CH + r) * NB + g] * ck2_spline[(c * R_CH + r) * NB + g];
        }
        acc2 += o2;
        __syncthreads();                      // red[] reused next source
    }
    const float att = sigmoidf_(acc2);
    ca_out[tid]  = att;
    caw_out[tid] = att * conv_w[c];
}

// ---------------------------------------------------------------------------
// Kernel 3: fused channel-scale + 1x1x1 conv reduce + spatial KAN + output.
// TDM loads the 128x128 fp32 tile into LDS once; WMMA f32 16x16x4 does the
// channel dot product; per-voxel spline is a 3-FMA Horner on precomputed
// segment coefficients.
// ---------------------------------------------------------------------------
__global__ void kbam_spatial_kernel(const float* __restrict__ x,
                                    const float* __restrict__ ca,        // [2,128]
                                    const float* __restrict__ caw,       // [2,128]
                                    const float* __restrict__ sk_base,   // [1]
                                    const float4* __restrict__ coeff,    // [11]
                                    float* __restrict__ out) {
    __shared__ float tile[C_CH * TILE_V];     // [c][v], 64 KB
    const int blk   = blockIdx.x;             // 0..4095
    const int b     = blk >> 11;              // / (HWD/TILE_V)
    const int t     = blk & 2047;
    const int vbase = t * TILE_V;
    const float* xb = x + (size_t)b * C_CH * HWD + vbase;

    // ---- TDM bulk load: 2D tile [dim1=128 rows][dim0=128 floats], stride HWD
    if (threadIdx.x < 32) {                   // wave 0 only (wave32)
        const uint64_t gaddr   = (uint64_t)(uintptr_t)xb;
        const uint32_t lds_off = (uint32_t)(uintptr_t)tile;  // low 32b = LDS offset
        u32x4 g0;
        g0[0] = 1u;                                             // count=1, user mode
        g0[1] = lds_off;                                        // lds_addr
        g0[2] = (uint32_t)gaddr;                                // global_addr[31:0]
        g0[3] = ((uint32_t)(gaddr >> 32) & 0x01FFFFFFu)
              | (2u << 30);                                     // type=2 ("image")
        i32x8 g1;
        g1[0] = (2 << 16);                                      // data_size=4B
        g1[1] = (int)((unsigned)TILE_V << 16);                  // tensor_dim0[15:0]
        g1[2] = (int)((unsigned)C_CH << 16);                    // tensor_dim1[15:0]
        g1[3] = (int)((unsigned)TILE_V << 16);                  // tile_dim0
        g1[4] = C_CH;                                           // tile_dim1 (tile_dim2=0)
        g1[5] = (int)HWD;                                       // tensor_dim0_stride[31:0]
        g1[6] = 0;
        g1[7] = 0;
        i32x4 z4 = {0, 0, 0, 0};
#if __clang_major__ >= 23
        i32x8 z8 = {0, 0, 0, 0, 0, 0, 0, 0};
        __builtin_amdgcn_tensor_load_to_lds(g0, g1, z4, z4, z8, 0);
#else
        __builtin_amdgcn_tensor_load_to_lds(g0, g1, z4, z4, 0);
#endif
        __builtin_amdgcn_s_wait_tensorcnt(0);
    }
    __syncthreads();

    // ---- Phase A: s[v] = sum_c caw[b,c] * tile[c][v] via WMMA f32 16x16x4
    const int lane = threadIdx.x & 31;
    const int wv   = threadIdx.x >> 5;
    const int hi   = (lane >> 4) & 1;         // lanes 16-31 carry K=2,3
    const int nA   = wv * 32 + (lane & 15);   // voxel group A column
    const int nB   = nA + 16;                 // voxel group B column
    const float* cawb = caw + b * C_CH;

    v8f accA = {0.f, 0.f, 0.f, 0.f, 0.f, 0.f, 0.f, 0.f};
    v8f accB = {0.f, 0.f, 0.f, 0.f, 0.f, 0.f, 0.f, 0.f};
#pragma unroll 4
    for (int k0 = 0; k0 < C_CH; k0 += 4) {
        const int kk = k0 + (hi ? 2 : 0);
        v2f a;  a[0]  = cawb[kk];                   a[1]  = cawb[kk + 1];
        v2f bA; bA[0] = tile[kk * TILE_V + nA];     bA[1] = tile[(kk + 1) * TILE_V + nA];
        v2f bB; bB[0] = tile[kk * TILE_V + nB];     bB[1] = tile[(kk + 1) * TILE_V + nB];
        accA = __builtin_amdgcn_wmma_f32_16x16x4_f32(false, a, false, bA,
                                                     (short)0, accA, false, false);
        accB = __builtin_amdgcn_wmma_f32_16x16x4_f32(false, a, false, bB,
                                                     (short)0, accB, false, false);
    }
    const float s = (lane < 16) ? accA[0] : accB[0];  // lane's own voxel sum

    // ---- Phase B: spatial attention, piecewise-cubic spline + fast sigmoid
    float tt = (s + 2.2f) * 2.5f;                     // (s - t_0)/h
    tt = fminf(fmaxf(tt, -1.0f), 12.0f);              // keep int conversion safe
    const float fm = floorf(tt);
    const int   m  = (int)fm;
    const bool  valid = (m >= 0) && (m <= 10);
    const int   mc = min(max(m, 0), 10);
    const float4 cf = coeff[mc];                      // cached b128 load
    const float u  = tt - fm;
    float sp = ((cf.w * u + cf.z) * u + cf.y) * u + cf.x;
    sp = valid ? sp : 0.0f;
    const float satt = sigmoidf_(siluf_(s) * sk_base[0] + sp);

    // ---- Phase C: out[b,c,v] = x * ca[b,c] * satt   (tile reread from LDS)
    const float* cab = ca + b * C_CH;
    const int v = threadIdx.x;
    float* ob = out + (size_t)b * C_CH * HWD + vbase + v;
#pragma unroll 4
    for (int c = 0; c < C_CH; ++c)
        ob[(size_t)c * HWD] = tile[c * TILE_V + v] * cab[c] * satt;
}

// ---------------------------------------------------------------------------
extern "C" void kernel_launch(void* const* d_in, const int* in_sizes, int n_in,
                              void* d_out, int out_size, void* d_ws, size_t ws_size,
                              hipStream_t stream) {
    const float* x          = (const float*)d_in[0];
    const float* ck1_base   = (const float*)d_in[1];
    const float* ck1_spline = (const float*)d_in[2];
    const float* ck2_base   = (const float*)d_in[3];
    const float* ck2_spline = (const float*)d_in[4];
    const float* conv_w     = (const float*)d_in[5];
    const float* sk_base    = (const float*)d_in[6];
    const float* sk_spline  = (const float*)d_in[7];
    float* out = (float*)d_out;

    float* ws    = (float*)d_ws;
    float* yavg  = ws;            // 256
    float* ymax  = ws + 256;      // 256
    float* ca    = ws + 512;      // 256
    float* caw   = ws + 768;      // 256
    float4* coef = (float4*)(ws + 1024);  // 11 x float4 (16B aligned)

    kbam_pool_kernel<<<2 * C_CH, 256, 0, stream>>>(x, yavg, ymax);
    kbam_chatt_kernel<<<1, 256, 0, stream>>>(yavg, ymax, ck1_base, ck1_spline,
                                             ck2_base, ck2_spline, conv_w,
                                             sk_spline, ca, caw, coef);
    kbam_spatial_kernel<<<2 * (HWD / TILE_V), TILE_V, 0, stream>>>(
        x, ca, caw, sk_base, coef, out);
}